// RecognizerLoss_72060961292714
// MI455X (gfx1250) — compile-verified
//
#include <hip/hip_runtime.h>
#include <cstdint>
#include <cstddef>

// ---------------------------------------------------------------------------
// CTC loss forward, one wave32 per batch sample (N=128 waves total).
//
//   log_probs [T, N, C=128] fp32 (log-softmaxed), targets [N, S], blank=0
//   L = 2S+1 = 257 lattice states live entirely in registers:
//     lane l holds states 8l..8l+7 (a0..a7); lane 31 additionally state 256 (a8)
//   s-1 / s-2 neighbors cross lanes only at lane boundaries -> one
//   __shfl_up pair per step, NO workgroup barriers on the 1023-step chain.
//
//   log_probs rows are streamed global->LDS with CDNA5 async copies:
//   one global_load_async_to_lds_b128 per row (32 lanes x 16B = 512B row),
//   16 LDS buffers, 14 rows in flight (ASYNCcnt pipeline, deep enough to
//   cover HBM latency vs the ~100-cycle step time).
// ---------------------------------------------------------------------------

#define NEGF  (-1.0e30f)
#define CMAX  128                 // classes (fixed by the reference)
#define SMAX  128
#define LMAX  (2 * SMAX + 1)      // 257
#define DBUF  16                  // LDS row buffers (power of two)

__device__ __forceinline__ void async_row_b128(uint32_t lds_byte_off, const float* gp) {
  // Each lane copies 16 bytes from its global address to its LDS offset.
  asm volatile("global_load_async_to_lds_b128 %0, %1, off"
               :: "v"(lds_byte_off), "v"((uint64_t)(uintptr_t)gp)
               : "memory");
}
#define WAIT_ASYNC(n) asm volatile("s_wait_asynccnt " #n ::: "memory")

__device__ __forceinline__ float lse2(float x, float y) {
  float m = fmaxf(x, y);
  return m + __logf(__expf(x - m) + __expf(y - m));
}
__device__ __forceinline__ float lse3(float x, float y, float z) {
  float m = fmaxf(x, fmaxf(y, z));
  return m + __logf(__expf(x - m) + __expf(y - m) + __expf(z - m));
}

extern "C" __global__ void __launch_bounds__(32)
ctc_alpha_wave(const float* __restrict__ log_probs,
               const int*   __restrict__ targets,
               const int*   __restrict__ input_lengths,
               const int*   __restrict__ target_lengths,
               float*       __restrict__ losses,
               int T, int N, int C, int S)
{
  const int lane = threadIdx.x;   // 0..31
  const int n    = blockIdx.x;

  __shared__ float s_lp[DBUF][CMAX];   // streamed log_probs rows (16 x 512 B)
  __shared__ float s_out[LMAX];        // final alphas for the readout

  const float* rowbase = log_probs + (size_t)n * (size_t)C + (size_t)(lane * 4);
  const size_t tstride = (size_t)N * (size_t)C;
  const uint32_t ldsbase = (uint32_t)(uintptr_t)&s_lp[0][0] + (uint32_t)(lane * 16);

  // ---- prologue: put rows 0..13 in flight --------------------------------
  for (int r = 0; r < 14; ++r)
    if (r < T)
      async_row_b128(ldsbase + (uint32_t)r * (CMAX * 4), rowbase + (size_t)r * tstride);

  // ---- per-lane lattice metadata (4 odd states per lane) -----------------
  const int* tgt = targets + (size_t)n * (size_t)S;
  const int ti = 4 * lane;                     // first target index of this lane
  const int t0 = tgt[ti],     t1 = tgt[ti + 1];
  const int t2 = tgt[ti + 2], t3 = tgt[ti + 3];
  const int tp = (ti > 0) ? tgt[ti - 1] : 0;
  const bool sk0 = (ti > 0) && (t0 != tp);     // state 8l+1 (s=1 never skips)
  const bool sk1 = (t1 != t0);                 // state 8l+3
  const bool sk2 = (t2 != t1);                 // state 8l+5
  const bool sk3 = (t3 != t2);                 // state 8l+7

  if (T >= 15) { WAIT_ASYNC(13); } else { WAIT_ASYNC(0); }   // row 0 landed

  // ---- alpha at t=0 ------------------------------------------------------
  float a0 = NEGF, a1 = NEGF, a2 = NEGF, a3 = NEGF, a4 = NEGF,
        a5 = NEGF, a6 = NEGF, a7 = NEGF, a8 = NEGF;
  if (lane == 0) {
    a0 = s_lp[0][0];        // state 0: blank
    a1 = s_lp[0][t0];       // state 1: first label
  }

  const int ilen = input_lengths[n];

  auto step = [&](int t) {
    const float* lp = &s_lp[t & (DBUF - 1)][0];
    // neighbor lane's old alpha[8l-1], alpha[8l-2]
    float P1 = __shfl_up(a7, 1);
    float P2 = __shfl_up(a6, 1);
    if (lane == 0) { P1 = NEGF; P2 = NEGF; }
    const float lpb = lp[0];                   // blank emission (broadcast)
    const float g0 = lp[t0], g1 = lp[t1], g2 = lp[t2], g3 = lp[t3];
    // even states: skip impossible; odd states: skip iff labels differ
    float n0 = lse2(a0, P1)                      + lpb;
    float n1 = lse3(a1, a0, sk0 ? P1 : NEGF)     + g0;
    float n2 = lse2(a2, a1)                      + lpb;
    float n3 = lse3(a3, a2, sk1 ? a1 : NEGF)     + g1;
    float n4 = lse2(a4, a3)                      + lpb;
    float n5 = lse3(a5, a4, sk2 ? a3 : NEGF)     + g2;
    float n6 = lse2(a6, a5)                      + lpb;
    float n7 = lse3(a7, a6, sk3 ? a5 : NEGF)     + g3;
    float n8 = lse2(a8, a7)                      + lpb;   // state 256 (lane 31)
    (void)P2;
    if (t < ilen) {                             // freeze past input_length
      a0 = n0; a1 = n1; a2 = n2; a3 = n3; a4 = n4;
      a5 = n5; a6 = n6; a7 = n7; a8 = n8;
    }
  };

  // ---- steady state: 14 rows in flight, wait for exactly row t -----------
  for (int t = 1; t + 13 < T; ++t) {
    const int r = t + 13;                      // slot (t+13)&15 == (t-3)&15: free
    async_row_b128(ldsbase + (uint32_t)((r & (DBUF - 1)) * (CMAX * 4)),
                   rowbase + (size_t)r * tstride);
    WAIT_ASYNC(13);                            // oldest (row t) complete
    step(t);
  }
  // ---- epilogue: everything already issued, drain then consume -----------
  for (int t = (T - 13 > 1 ? T - 13 : 1); t < T; ++t) {
    WAIT_ASYNC(0);
    step(t);
  }

  // ---- readout -----------------------------------------------------------
  s_out[8 * lane + 0] = a0; s_out[8 * lane + 1] = a1;
  s_out[8 * lane + 2] = a2; s_out[8 * lane + 3] = a3;
  s_out[8 * lane + 4] = a4; s_out[8 * lane + 5] = a5;
  s_out[8 * lane + 6] = a6; s_out[8 * lane + 7] = a7;
  if (lane == 31) s_out[256] = a8;
  __syncthreads();                             // single wave: compiler fence
  if (lane == 0) {
    const int tl = target_lengths[n];
    const float ll = lse2(s_out[2 * tl], s_out[2 * tl - 1]);
    losses[n] = -ll / (float)tl;
  }
}

// Deterministic fixed-order mean reduction (no float atomics).
extern "C" __global__ void ctc_reduce_kernel(const float* __restrict__ losses,
                                             float* __restrict__ out, int N)
{
  if (threadIdx.x == 0 && blockIdx.x == 0) {
    float acc = 0.0f;
    for (int i = 0; i < N; ++i) acc += losses[i];
    out[0] = acc / (float)N;
  }
}

extern "C" void kernel_launch(void* const* d_in, const int* in_sizes, int n_in,
                              void* d_out, int out_size, void* d_ws, size_t ws_size,
                              hipStream_t stream) {
  const float* log_probs      = (const float*)d_in[0];
  const int*   targets        = (const int*)d_in[1];
  const int*   input_lengths  = (const int*)d_in[2];
  const int*   target_lengths = (const int*)d_in[3];

  const int N = in_sizes[2];                 // 128
  const int S = in_sizes[1] / N;             // 128
  const int C = CMAX;                        // 128 (fixed by reference)
  const int T = in_sizes[0] / (N * C);       // 1024

  float* losses = (float*)d_ws;              // N floats of scratch

  ctc_alpha_wave<<<N, 32, 0, stream>>>(
      log_probs, targets, input_lengths, target_lengths, losses, T, N, C, S);
  ctc_reduce_kernel<<<1, 32, 0, stream>>>(losses, (float*)d_out, N);
}